// Transfer1_0_73332271612004
// MI455X (gfx1250) — compile-verified
//
#include <hip/hip_runtime.h>

typedef float v2f __attribute__((ext_vector_type(2)));
typedef float v8f __attribute__((ext_vector_type(8)));

#define H      128
#define N_X    250000
#define N_SRC  50000
#define N_Y    50000
#define N_INT  500000
#define E_NODE 1000000
#define E_DOM  500000
#define BN_EPS 1e-5f

__device__ __forceinline__ int lower_bound_i(const int* __restrict__ a, int n, int key) {
    int l = 0, r = n;
    while (l < r) {
        int m = (l + r) >> 1;
        if (a[m] < key) l = m + 1; else r = m;
    }
    return l;
}

// xs[s][c] = sum over rows r with domain_indicator[r]==s of x[r][c]   (indicator sorted)
__global__ void seg_sum_x_kernel(const float* __restrict__ x, const int* __restrict__ ind,
                                 float* __restrict__ out) {
    int s = blockIdx.x;          // segment id  [0, N_SRC)
    int c = threadIdx.x;         // column      [0, H)
    int lo = lower_bound_i(ind, N_X, s);
    int hi = lower_bound_i(ind, N_X, s + 1);
    float acc = 0.0f;
    for (int r = lo; r < hi; ++r) acc += x[(size_t)r * H + c];
    out[(size_t)s * H + c] = acc;
}

// xint[i][c] = sum over edges e with intersect_indicator[e]==i of x[nm0[e]][c]  (indicator sorted)
__global__ void seg_sum_xint_kernel(const float* __restrict__ x, const int* __restrict__ nm0,
                                    const int* __restrict__ ind, float* __restrict__ out) {
    int i = blockIdx.x;          // intersection id [0, N_INT)
    int c = threadIdx.x;
    int lo = lower_bound_i(ind, E_NODE, i);
    int hi = lower_bound_i(ind, E_NODE, i + 1);
    float acc = 0.0f;
    for (int e = lo; e < hi; ++e) {
        int r = nm0[e];
        acc += x[(size_t)r * H + c];
    }
    out[(size_t)i * H + c] = acc;
}

// Shared WMMA inner step: batch all 8 B-fragment loads into live registers,
// then issue 8 independent WMMAs (one wait instead of 8 serialized ones).
__device__ __forceinline__ void wmma_kstep(const float* __restrict__ W, const float* arow,
                                           int k, int l16, v8f c[8]) {
    v2f a = *(const v2f*)(arow + k);
    v2f b[8];
#pragma unroll
    for (int nt = 0; nt < 8; ++nt)
        b[nt] = *(const v2f*)(W + (nt * 16 + l16) * H + k);
#pragma unroll
    for (int nt = 0; nt < 8; ++nt)
        c[nt] = __builtin_amdgcn_wmma_f32_16x16x4_f32(
            false, a, false, b[nt], (short)0, c[nt], false, false);
}

// out = A @ W.T  (rows = mtiles*16, K = N = 128). One wave per 16-row strip.
// A f32 16x4 frag: lane (l16,half) vgpr v -> A[row0+l16][kb*4 + 2*half + v]
// B f32 4x16 frag: lane (l16,half) vgpr v -> W[nt*16+l16][kb*4 + 2*half + v]   (B[k][n] = W[n][k])
// C:              vgpr j, lane (l16,half) -> out[row0 + 8*half + j][nt*16 + l16]
// NOTE: A and out may alias (in-place) — c depends on all A loads, stores follow; no restrict.
__global__ void wmma_rowlin_kernel(const float* A, const float* __restrict__ W,
                                   float* out, int mtiles) {
    int lane = threadIdx.x & 31;
    int wave = threadIdx.x >> 5;
    int tile = blockIdx.x * 4 + wave;
    if (tile >= mtiles) return;                 // wave-uniform: EXEC stays all-ones
    int half = lane >> 4;
    int l16  = lane & 15;
    int row0 = tile * 16;
    const float* arow = A + (size_t)(row0 + l16) * H;

    v8f c[8] = {};
    for (int kb = 0; kb < 32; ++kb)
        wmma_kstep(W, arow, kb * 4 + half * 2, l16, c);

#pragma unroll
    for (int nt = 0; nt < 8; ++nt)
#pragma unroll
        for (int j = 0; j < 8; ++j)
            out[(size_t)(row0 + half * 8 + j) * H + nt * 16 + l16] = c[nt][j];
}

// Fused: msg = (xint @ Wxi.T) + xs_lin[dm0] + yt[dm1]; write msg in-place over xint;
// accumulate per-column sum / sum-of-squares into stats[0..H) / stats[H..2H) for BatchNorm.
__global__ void msg_pass1_kernel(float* xint, const float* __restrict__ Wxi,
                                 const float* __restrict__ xs, const float* __restrict__ yt,
                                 const int* __restrict__ dm0, const int* __restrict__ dm1,
                                 float* __restrict__ stats, int mtiles) {
    __shared__ float lsum[H];
    __shared__ float lsq[H];
    int tid = threadIdx.x;       // blockDim.x == 128
    lsum[tid] = 0.0f;
    lsq[tid]  = 0.0f;
    __syncthreads();

    int lane = tid & 31;
    int wave = tid >> 5;
    int tile = blockIdx.x * 4 + wave;
    if (tile < mtiles) {                        // wave-uniform guard
        int half = lane >> 4;
        int l16  = lane & 15;
        int row0 = tile * 16;
        const float* arow = xint + (size_t)(row0 + l16) * H;

        v8f c[8] = {};
        for (int kb = 0; kb < 32; ++kb)
            wmma_kstep(Wxi, arow, kb * 4 + half * 2, l16, c);

        int d0[8], d1[8];
#pragma unroll
        for (int j = 0; j < 8; ++j) {
            int e = row0 + half * 8 + j;
            d0[j] = dm0[e];
            d1[j] = dm1[e];
        }
#pragma unroll
        for (int nt = 0; nt < 8; ++nt) {
            int col = nt * 16 + l16;
            float s = 0.0f, q = 0.0f;
#pragma unroll
            for (int j = 0; j < 8; ++j) {
                int e = row0 + half * 8 + j;
                float v = c[nt][j]
                        + xs[(size_t)d0[j] * H + col]
                        + yt[(size_t)d1[j] * H + col];
                xint[(size_t)e * H + col] = v;  // in-place msg (wave owns these rows)
                s += v;
                q += v * v;
            }
            atomicAdd(&lsum[col], s);
            atomicAdd(&lsq[col], q);
        }
    }
    __syncthreads();
    atomicAdd(&stats[tid],     lsum[tid]);
    atomicAdd(&stats[H + tid], lsq[tid]);
}

// Convert (sum, sumsq) -> (scale, shift) for fused BN: out = msg*scale + shift, then ReLU.
__global__ void bn_finalize_kernel(float* __restrict__ stats,
                                   const float* __restrict__ bn_w,
                                   const float* __restrict__ bn_b) {
    int c = threadIdx.x;         // one block of H threads
    float inv_n = 1.0f / (float)E_DOM;
    float mean  = stats[c] * inv_n;
    float var   = fmaxf(stats[H + c] * inv_n - mean * mean, 0.0f);
    float scale = bn_w[c] * rsqrtf(var + BN_EPS);
    float shift = bn_b[c] - mean * scale;
    stats[c]     = scale;
    stats[H + c] = shift;
}

// out[dm1[e]][c] += relu(msg[e][c]*scale[c] + shift[c])
__global__ void scatter_kernel(const float* __restrict__ msg, const int* __restrict__ dm1,
                               const float* __restrict__ stats, float* __restrict__ out) {
    long long idx = (long long)blockIdx.x * blockDim.x + threadIdx.x;
    if (idx >= (long long)E_DOM * H) return;
    int c = (int)(idx & (H - 1));
    int e = (int)(idx >> 7);
    float v = msg[idx] * stats[c] + stats[H + c];
    v = fmaxf(v, 0.0f);
    atomicAdd(&out[(size_t)dm1[e] * H + c], v);
}

extern "C" void kernel_launch(void* const* d_in, const int* in_sizes, int n_in,
                              void* d_out, int out_size, void* d_ws, size_t ws_size,
                              hipStream_t stream) {
    const float* x       = (const float*)d_in[0];
    const float* y       = (const float*)d_in[1];
    const int*   dom_ind = (const int*)d_in[2];
    const int*   nm      = (const int*)d_in[3];   // [2, E_NODE]
    const int*   int_ind = (const int*)d_in[4];
    const int*   dm      = (const int*)d_in[5];   // [2, E_DOM]
    const float* Wxs     = (const float*)d_in[6];
    const float* Wxi     = (const float*)d_in[7];
    const float* Wy      = (const float*)d_in[8];
    const float* bn_w    = (const float*)d_in[9];
    const float* bn_b    = (const float*)d_in[10];
    const int* nm0 = nm;
    const int* dm0 = dm;
    const int* dm1 = dm + E_DOM;

    float* out_f = (float*)d_out;

    // workspace layout (floats): xs[N_SRC*H] | yt[N_Y*H] | xint[N_INT*H] | stats[2*H]
    float* ws    = (float*)d_ws;
    float* xs    = ws;
    float* yt    = xs + (size_t)N_SRC * H;
    float* xint  = yt + (size_t)N_Y * H;
    float* stats = xint + (size_t)N_INT * H;

    (void)hipMemsetAsync(d_out, 0, (size_t)out_size * sizeof(float), stream);
    (void)hipMemsetAsync(stats, 0, 2 * H * sizeof(float), stream);

    seg_sum_x_kernel<<<N_SRC, H, 0, stream>>>(x, dom_ind, xs);
    seg_sum_xint_kernel<<<N_INT, H, 0, stream>>>(x, nm0, int_ind, xint);

    int mt_small = N_SRC / 16;                       // 3125
    wmma_rowlin_kernel<<<(mt_small + 3) / 4, 128, 0, stream>>>(xs, Wxs, xs, mt_small);  // in-place
    wmma_rowlin_kernel<<<(mt_small + 3) / 4, 128, 0, stream>>>(y,  Wy,  yt, mt_small);

    int mt_big = N_INT / 16;                         // 31250
    msg_pass1_kernel<<<(mt_big + 3) / 4, 128, 0, stream>>>(xint, Wxi, xs, yt, dm0, dm1, stats, mt_big);

    bn_finalize_kernel<<<1, H, 0, stream>>>(stats, bn_w, bn_b);

    long long total = (long long)E_DOM * H;
    scatter_kernel<<<(unsigned)((total + 255) / 256), 256, 0, stream>>>(xint, dm1, stats, out_f);
}